// PlanNetwork_46042049413505
// MI455X (gfx1250) — compile-verified
//
#include <hip/hip_runtime.h>
#include <math.h>

// ---------------------------------------------------------------------------
// PlanNetwork forward for gfx1250 (MI455X).
//
// All dense math (QKV, O-proj, FFN1, FFN2, QK^T, P@V) runs on the WMMA f32
// pipe (V_WMMA_F32_16X16X4_F32) with fp32 accumulate to match reference
// numerics. HIDDEN=249 is padded to 256; attention heads (dim 83) are padded
// to 96 (Q/K/V/attn-out rows are 3*96 = 288 wide) so attention tiles map to
// 16x16 WMMA. Padded weight copies are built in workspace every call.
//
// Input index mapping assumption: d_in follows setup_inputs() dict order at
// top level (attn_bias, x, heights, params) with `params` flattened as a JAX
// pytree (dict keys recursively sorted):
//   3 columnEmbed, 4 fln_b, 5 fln_g, 6 heightEmbed,
//   7+16*l + {0 f1.b, 1 f1.w, 2 f2.b, 3 f2.w, 4 k.b, 5 k.w, 6 ln1_b, 7 ln1_g,
//             8 ln2_b, 9 ln2_g, 10 o.b, 11 o.w, 12 q.b, 13 q.w, 14 v.b, 15 v.w},
//   71 linFilter.b, 72 linFilter.w, 73 linFilter2.b, 74 linFilter2.w,
//   75 linJoin1.b, 76 linJoin1.w, 77 linJoin2.b, 78 linJoin2.w,
//   79 opEmbed, 80 posEmbed, 81 project.b, 82 project.w, 83 tableEmbed,
//   84 typeEmbed.   (n_in = 85)
// ---------------------------------------------------------------------------

typedef __attribute__((ext_vector_type(2))) float v2f;
typedef __attribute__((ext_vector_type(8))) float v8f;

#define Bsz   64
#define Nseq  512
#define HPAD  256      // padded HIDDEN (h buffer stride)
#define HID   249
#define ATTD  83
#define NHEAD 3
#define HDP   96       // padded head dim
#define QSTR  288      // q/k/v/y/attn-out row stride (3*96)
#define FFND  1024
#define NTOK  (Bsz * Nseq)       // 32768
#define SCALE_ATT 0.10976425998969035f  // 83^-0.5

__device__ __forceinline__ float leaky_f(float x) { return x >= 0.f ? x : 0.01f * x; }
__device__ __forceinline__ float gelu_f(float x) {
  return 0.5f * x * (1.0f + erff(x * 0.70710678118654752f));
}

__device__ __forceinline__ v8f wmma4(v2f a, v2f b, v8f c) {
  // D = A(16x4,f32) * B(4x16,f32) + C(16x16,f32)
  return __builtin_amdgcn_wmma_f32_16x16x4_f32(false, a, false, b, (short)0, c,
                                               false, false);
}

// ---------------------------------------------------------------------------
// Pad one layer's weights/biases into workspace.
// Per-layer float layout:
//   WQ 0       (256x288, head-padded cols)   WK 73728    WV 147456
//   WO 221184  (288x256, head-padded rows)
//   W1 294912  (256x1024)   W2 557056 (1024x256)
//   bq 819200 (288)  bk 819488  bv 819776  bo 820064 (256)  b2 820320 (256)
// Total 820576 floats -> grid 3206 x 256.
// ---------------------------------------------------------------------------
__global__ __launch_bounds__(256)
void pad_weights_kernel(const float* __restrict__ qw, const float* __restrict__ kw,
                        const float* __restrict__ vw, const float* __restrict__ ow,
                        const float* __restrict__ f1w, const float* __restrict__ f2w,
                        const float* __restrict__ qb, const float* __restrict__ kb,
                        const float* __restrict__ vb, const float* __restrict__ ob,
                        const float* __restrict__ f2b, float* __restrict__ dst) {
  int i = blockIdx.x * 256 + threadIdx.x;
  if (i >= 820576) return;
  float v;
  if (i < 221184) {                       // WQ/WK/WV: 256 x 288
    int which = i / 73728; int j = i % 73728; int r = j / QSTR; int c = j % QSTR;
    int head = c / HDP, off = c % HDP;
    const float* s = (which == 0) ? qw : (which == 1) ? kw : vw;
    v = (r < HID && off < ATTD) ? s[r * HID + head * ATTD + off] : 0.0f;
  } else if (i < 294912) {                // WO: 288 x 256
    int j = i - 221184; int r = j >> 8; int c = j & 255;
    int head = r / HDP, off = r % HDP;
    v = (off < ATTD && c < HID) ? ow[(head * ATTD + off) * HID + c] : 0.0f;
  } else if (i < 557056) {                // W1 256x1024
    int j = i - 294912; int r = j >> 10; int c = j & 1023;
    v = (r < HID) ? f1w[r * FFND + c] : 0.0f;
  } else if (i < 819200) {                // W2 1024x256
    int j = i - 557056; int r = j >> 8; int c = j & 255;
    v = (c < HID) ? f2w[r * HID + c] : 0.0f;
  } else if (i < 820064) {                // bq/bk/bv: 288 each
    int j = i - 819200; int which = j / QSTR; int c = j % QSTR;
    int head = c / HDP, off = c % HDP;
    const float* s = (which == 0) ? qb : (which == 1) ? kb : vb;
    v = (off < ATTD) ? s[head * ATTD + off] : 0.0f;
  } else {                                // bo, b2: 256 each
    int j = i - 820064; int which = j >> 8; int c = j & 255;
    const float* s = (which == 0) ? ob : f2b;
    v = (c < HID) ? s[c] : 0.0f;
  }
  dst[i] = v;
}

// ---------------------------------------------------------------------------
// Per-token feature embedding. One 128-thread block per token.
// ---------------------------------------------------------------------------
__global__ __launch_bounds__(128)
void embed_kernel(const float* __restrict__ X, const int* __restrict__ Hts,
                  const float* __restrict__ colE, const float* __restrict__ opE,
                  const float* __restrict__ posE, const float* __restrict__ typeE,
                  const float* __restrict__ tabE, const float* __restrict__ heE,
                  const float* __restrict__ WF, const float* __restrict__ bF,
                  const float* __restrict__ WF2, const float* __restrict__ bF2,
                  const float* __restrict__ WJ1, const float* __restrict__ bJ1,
                  const float* __restrict__ WJ2, const float* __restrict__ bJ2,
                  const float* __restrict__ WP, const float* __restrict__ bP,
                  float* __restrict__ Hout) {
  __shared__ float sx[20];
  __shared__ float s_in[256];
  __shared__ float s_t[96];
  __shared__ float s_filt[69];
  __shared__ float s_j[96];
  const int t = threadIdx.x;
  const int tok = blockIdx.x;
  if (t < 19) sx[t] = X[(size_t)tok * 19 + t];
  if (t < 69) s_filt[t] = 0.0f;
  __syncthreads();
  for (int i = 0; i < 3; ++i) {
    int ci = (int)sx[5 + i]; int oi = (int)sx[8 + i];
    if (t < 69)
      s_in[t] = (t < 64) ? colE[ci * 64 + t]
              : (t < 68) ? opE[oi * 4 + (t - 64)] : sx[11 + i];
    __syncthreads();
    if (t < 69) {
      float a = bF[t];
      for (int u = 0; u < 69; ++u) a += s_in[u] * WF[u * 69 + t];
      s_t[t] = leaky_f(a);
    }
    __syncthreads();
    if (t < 69) {
      float a = bF2[t];
      for (int u = 0; u < 69; ++u) a += s_t[u] * WF2[u * 69 + t];
      if (sx[14 + i] != 0.0f) s_filt[t] += leaky_f(a);
    }
    __syncthreads();
  }
  float denom = sx[14] + sx[15] + sx[16] + 1e-10f;
  if (t < 69) s_filt[t] /= denom;
  for (int j = t; j < 256; j += 128) {
    int m = j >> 6; int c = j & 63;
    int ji = (int)sx[1 + m];
    s_in[j] = colE[ji * 64 + c];
  }
  __syncthreads();
  if (t < 96) {
    float a = bJ1[t];
    for (int u = 0; u < 256; ++u) a += s_in[u] * WJ1[u * 96 + t];
    s_t[t] = leaky_f(a);
  }
  __syncthreads();
  if (t < 96) {
    float a = bJ2[t];
    for (int u = 0; u < 96; ++u) a += s_t[u] * WJ2[u * 96 + t];
    s_j[t] = leaky_f(a);
  }
  __syncthreads();
  int ti = (int)sx[0]; int tb = (int)sx[18]; int pi = (int)sx[17];
  for (int j = t; j < 233; j += 128) {
    float v;
    if (j < 32)       v = typeE[ti * 32 + j];
    else if (j < 101) v = s_filt[j - 32];
    else if (j < 197) v = s_j[j - 101];
    else if (j < 229) v = tabE[tb * 32 + (j - 197)];
    else              v = posE[pi * 4 + (j - 229)];
    s_in[j] = v;
  }
  __syncthreads();
  int hh = Hts[tok];
  for (int j = t; j < 256; j += 128) {
    float v;
    if (j < 233) {
      float a = bP[j];
      for (int u = 0; u < 233; ++u) a += s_in[u] * WP[u * 233 + j];
      v = leaky_f(a);
    } else if (j < HID) {
      v = heE[hh * 16 + (j - 233)];
    } else {
      v = 0.0f;  // keep pad columns zero
    }
    Hout[(size_t)tok * HPAD + j] = v;
  }
}

// ---------------------------------------------------------------------------
// LayerNorm: reads H (stride 256, 249 valid), writes Y (stride 288) with
// zeros in cols 249..287. 1 wave per token.
// ---------------------------------------------------------------------------
__global__ __launch_bounds__(256)
void ln_kernel(const float* __restrict__ X, float* __restrict__ Yd,
               const float* __restrict__ g, const float* __restrict__ bb) {
  const int lane = threadIdx.x & 31, wave = threadIdx.x >> 5;
  const int tok = blockIdx.x * 8 + wave;
  const float* xr = X + (size_t)tok * HPAD;
  float v[9]; float s = 0.f;
  for (int i = 0; i < 9; ++i) {
    int j = lane + 32 * i;
    v[i] = (j < HPAD) ? xr[j] : 0.0f;
    if (j < HID) s += v[i];
  }
  for (int o = 16; o > 0; o >>= 1) s += __shfl_xor(s, o, 32);
  float mean = s / (float)HID;
  float q = 0.f;
  for (int i = 0; i < 9; ++i) {
    int j = lane + 32 * i;
    if (j < HID) { float d = v[i] - mean; q += d * d; }
  }
  for (int o = 16; o > 0; o >>= 1) q += __shfl_xor(q, o, 32);
  float rs = rsqrtf(q / (float)HID + 1e-5f);
  float* yr = Yd + (size_t)tok * QSTR;
  for (int i = 0; i < 9; ++i) {
    int j = lane + 32 * i;
    yr[j] = (j < HID) ? (v[i] - mean) * rs * g[j] + bb[j] : 0.0f;
  }
}

// Final LN: tokens (b, 0) only; writes d_out (64 x 249).
__global__ __launch_bounds__(256)
void final_ln_kernel(const float* __restrict__ X, float* __restrict__ out,
                     const float* __restrict__ g, const float* __restrict__ bb) {
  const int lane = threadIdx.x & 31, wave = threadIdx.x >> 5;
  const int b = blockIdx.x * 8 + wave;  // 0..63
  const float* xr = X + (size_t)b * Nseq * HPAD;
  float v[8]; float s = 0.f;
  for (int i = 0; i < 8; ++i) {
    int j = lane + 32 * i; v[i] = xr[j];
    if (j < HID) s += v[i];
  }
  for (int o = 16; o > 0; o >>= 1) s += __shfl_xor(s, o, 32);
  float mean = s / (float)HID;
  float q = 0.f;
  for (int i = 0; i < 8; ++i) {
    int j = lane + 32 * i;
    if (j < HID) { float d = v[i] - mean; q += d * d; }
  }
  for (int o = 16; o > 0; o >>= 1) q += __shfl_xor(q, o, 32);
  float rs = rsqrtf(q / (float)HID + 1e-5f);
  for (int i = 0; i < 8; ++i) {
    int j = lane + 32 * i;
    if (j < HID) out[(size_t)b * HID + j] = (v[i] - mean) * rs * g[j] + bb[j];
  }
}

// ---------------------------------------------------------------------------
// WMMA f32 GEMM: C[M x N] = A[M x K] * W[K x N] + bias (+ resid).
// Block = 4 waves, 128 rows x 32 cols; each wave a 32x32 tile (2x2 blocking).
// Weight tile (K x 32) staged in LDS quad-packed so a lane's B fragment is a
// single aligned ds_load_b64 (conflict-free for both lane halves).
// Grid: (N/32, M/128).  K <= 288, multiple of 4.
// ---------------------------------------------------------------------------
__global__ __launch_bounds__(128)
void gemm_kernel(const float* __restrict__ A, const float* __restrict__ W,
                 const float* __restrict__ bias, float* __restrict__ C,
                 const float* __restrict__ resid, int N, int K, int lda) {
  __shared__ __align__(16) float sW[72 * 128];  // up to K=288 -> 36 KB
  const int lane = threadIdx.x & 31, wave = threadIdx.x >> 5;
  const int half = lane >> 4, lrow = lane & 15;
  const int nTile = blockIdx.x * 32;
  const int mBase = blockIdx.y * 128;

  for (int idx = threadIdx.x; idx < (K >> 2) * 32; idx += 128) {
    int q = idx >> 5, c = idx & 31;            // k-quad, col
    const float* src = W + (size_t)(q * 4) * N + nTile + c;
    float* d = &sW[q * 128 + c * 4];
    d[0] = src[0];
    d[1] = src[N];
    d[2] = src[2 * N];
    d[3] = src[3 * N];
  }
  __syncthreads();

  const int r0 = mBase + wave * 32 + lrow;
  const float* A0 = A + (size_t)r0 * lda + 2 * half;
  const float* A1 = A0 + (size_t)16 * lda;
  v8f a00 = {}, a01 = {}, a10 = {}, a11 = {};
#pragma unroll 8
  for (int k = 0; k < K; k += 4) {
    v2f fa0 = *(const v2f*)(A0 + k);
    v2f fa1 = *(const v2f*)(A1 + k);
    const float* bp = &sW[(k >> 2) * 128 + lrow * 4 + 2 * half];
    v2f fb0 = *(const v2f*)bp;          // cols nTile+lrow
    v2f fb1 = *(const v2f*)(bp + 64);   // cols nTile+lrow+16
    a00 = wmma4(fa0, fb0, a00);
    a01 = wmma4(fa0, fb1, a01);
    a10 = wmma4(fa1, fb0, a10);
    a11 = wmma4(fa1, fb1, a11);
  }
  const int c0 = nTile + lrow, c1 = c0 + 16;
  const float bi0 = bias[c0], bi1 = bias[c1];
#pragma unroll
  for (int r = 0; r < 8; ++r) {
    int rowA = mBase + wave * 32 + r + 8 * half;
    int rowB = rowA + 16;
    size_t oA0 = (size_t)rowA * N + c0, oA1 = (size_t)rowA * N + c1;
    size_t oB0 = (size_t)rowB * N + c0, oB1 = (size_t)rowB * N + c1;
    float v00 = a00[r] + bi0, v01 = a01[r] + bi1;
    float v10 = a10[r] + bi0, v11 = a11[r] + bi1;
    if (resid) {
      v00 += resid[oA0]; v01 += resid[oA1];
      v10 += resid[oB0]; v11 += resid[oB1];
    }
    C[oA0] = v00; C[oA1] = v01;
    C[oB0] = v10; C[oB1] = v11;
  }
}

// ---------------------------------------------------------------------------
// Fused FFN: H += gelu(Y*W1 + b1) * W2 + b2, per 32-token tile. Y stride 288.
// Hidden in four 256-col quarters via a 32x260 LDS stage.
// ---------------------------------------------------------------------------
#define LROW 260
__global__ __launch_bounds__(256)
void ffn_kernel(float* __restrict__ H, const float* __restrict__ Y,
                const float* __restrict__ W1p, const float* __restrict__ bias1,
                const float* __restrict__ W2p, const float* __restrict__ bias2p) {
  __shared__ __align__(16) float s_hid[32 * LROW];
  const int lane = threadIdx.x & 31, wave = threadIdx.x >> 5;  // wave 0..7
  const int half = lane >> 4, lrow = lane & 15;
  const int mTile = blockIdx.x * 32;
  const int cw = wave * 32;
  const float* A0 = Y + (size_t)(mTile + lrow) * QSTR + 2 * half;
  const float* A1 = A0 + (size_t)16 * QSTR;
  v8f o00 = {}, o01 = {}, o10 = {}, o11 = {};
  for (int kq = 0; kq < 4; ++kq) {
    v8f h00 = {}, h01 = {}, h10 = {}, h11 = {};
    const float* Wc = W1p + kq * 256 + cw + lrow;
#pragma unroll 8
    for (int k = 0; k < HPAD; k += 4) {
      v2f fa0 = *(const v2f*)(A0 + k);
      v2f fa1 = *(const v2f*)(A1 + k);
      int r0 = k + 2 * half;
      v2f fb0, fb1;
      fb0.x = Wc[(size_t)r0 * FFND];      fb0.y = Wc[(size_t)(r0 + 1) * FFND];
      fb1.x = Wc[(size_t)r0 * FFND + 16]; fb1.y = Wc[(size_t)(r0 + 1) * FFND + 16];
      h00 = wmma4(fa0, fb0, h00);
      h01 = wmma4(fa0, fb1, h01);
      h10 = wmma4(fa1, fb0, h10);
      h11 = wmma4(fa1, fb1, h11);
    }
    {
      const int c0 = cw + lrow, c1 = c0 + 16;
      const float bb0 = bias1[kq * 256 + c0], bb1 = bias1[kq * 256 + c1];
#pragma unroll
      for (int r = 0; r < 8; ++r) {
        int row = r + 8 * half;
        s_hid[row * LROW + c0]        = gelu_f(h00[r] + bb0);
        s_hid[row * LROW + c1]        = gelu_f(h01[r] + bb1);
        s_hid[(row + 16) * LROW + c0] = gelu_f(h10[r] + bb0);
        s_hid[(row + 16) * LROW + c1] = gelu_f(h11[r] + bb1);
      }
    }
    __syncthreads();
    const float* Ar0 = &s_hid[lrow * LROW + 2 * half];
    const float* Ar1 = &s_hid[(lrow + 16) * LROW + 2 * half];
    const float* W2c = W2p + (size_t)(kq * 256) * HPAD + cw + lrow;
#pragma unroll 8
    for (int k = 0; k < 256; k += 4) {
      v2f fa0 = *(const v2f*)(Ar0 + k);
      v2f fa1 = *(const v2f*)(Ar1 + k);
      int r0 = k + 2 * half;
      v2f fb0, fb1;
      fb0.x = W2c[(size_t)r0 * HPAD];      fb0.y = W2c[(size_t)(r0 + 1) * HPAD];
      fb1.x = W2c[(size_t)r0 * HPAD + 16]; fb1.y = W2c[(size_t)(r0 + 1) * HPAD + 16];
      o00 = wmma4(fa0, fb0, o00);
      o01 = wmma4(fa0, fb1, o01);
      o10 = wmma4(fa1, fb0, o10);
      o11 = wmma4(fa1, fb1, o11);
    }
    __syncthreads();
  }
  const int c0 = cw + lrow, c1 = c0 + 16;
  const float bb0 = bias2p[c0], bb1 = bias2p[c1];
#pragma unroll
  for (int r = 0; r < 8; ++r) {
    int rowA = mTile + r + 8 * half;
    int rowB = rowA + 16;
    size_t oA0 = (size_t)rowA * HPAD + c0, oA1 = (size_t)rowA * HPAD + c1;
    size_t oB0 = (size_t)rowB * HPAD + c0, oB1 = (size_t)rowB * HPAD + c1;
    H[oA0] += o00[r] + bb0;
    H[oA1] += o01[r] + bb1;
    H[oB0] += o10[r] + bb0;
    H[oB1] += o11[r] + bb1;
  }
}

// ---------------------------------------------------------------------------
// WMMA attention. Block = 4 waves, one (batch, head, 16-query tile).
// Phase S: scores(16x512) = Q*K^T via WMMA, both fragments row-major v2f
//          loads; scaled by SCALE*bias, staged in LDS.
// Softmax over keys (16 rows x 8 chunks of 64).
// Phase O: out(16x96) = P @ V via WMMA (A from LDS, B strided V loads),
//          normalized by row sums at epilogue. Writes O (= Yb, stride 288).
// ---------------------------------------------------------------------------
#define LS 516
__global__ __launch_bounds__(128)
void attn_kernel(const float* __restrict__ Q, const float* __restrict__ Kb,
                 const float* __restrict__ Vb, const float* __restrict__ bias,
                 float* __restrict__ O) {
  __shared__ __align__(16) float s_s[16 * LS];
  __shared__ float s_red[16 * 8];
  __shared__ float s_inv[16];
  const int lane = threadIdx.x & 31, wave = threadIdx.x >> 5;
  const int half = lane >> 4, lrow = lane & 15;
  const int qt = blockIdx.x;        // 16-query tile, 0..31
  const int hd = blockIdx.y;        // head 0..2
  const int b  = blockIdx.z;        // batch 0..63
  const int q0 = qt * 16;
  const size_t base = (size_t)b * Nseq;
  const float* Qrow = Q + (base + q0 + lrow) * QSTR + hd * HDP + 2 * half;
  // ---- phase S ----
  for (int kt = wave * 8; kt < wave * 8 + 8; ++kt) {
    const int key0 = kt * 16;
    const float* Krow = Kb + (base + key0 + lrow) * QSTR + hd * HDP + 2 * half;
    v8f acc = {};
#pragma unroll
    for (int k = 0; k < HDP; k += 4) {
      v2f fa = *(const v2f*)(Qrow + k);
      v2f fb = *(const v2f*)(Krow + k);   // B = K^T fragment: K row-major
      acc = wmma4(fa, fb, acc);
    }
    const int key = key0 + lrow;
#pragma unroll
    for (int r = 0; r < 8; ++r) {
      int qr = r + 8 * half;
      float bv = bias[((size_t)b * 513 + (q0 + qr + 1)) * 513 + key + 1];
      s_s[qr * LS + key] = acc[r] * (SCALE_ATT * bv);
    }
  }
  __syncthreads();
  // ---- softmax over 512 keys per query row ----
  {
    const int row = threadIdx.x & 15, ch = threadIdx.x >> 4;  // 8 chunks x 64
    float m = -3.402823466e38f;
    for (int k = ch * 64; k < ch * 64 + 64; ++k)
      m = fmaxf(m, s_s[row * LS + k]);
    s_red[row * 8 + ch] = m;
    __syncthreads();
    float mx = s_red[row * 8];
    for (int i = 1; i < 8; ++i) mx = fmaxf(mx, s_red[row * 8 + i]);
    __syncthreads();
    float sm = 0.f;
    for (int k = ch * 64; k < ch * 64 + 64; ++k) {
      float e = expf(s_s[row * LS + k] - mx);
      s_s[row * LS + k] = e; sm += e;
    }
    s_red[row * 8 + ch] = sm;
    __syncthreads();
    if (ch == 0) {
      float tot = 0.f;
      for (int i = 0; i < 8; ++i) tot += s_red[row * 8 + i];
      s_inv[row] = 1.0f / tot;
    }
  }
  __syncthreads();
  // ---- phase O: out = P @ V ----
  const float* Ar = &s_s[lrow * LS + 2 * half];
  for (int tt = wave; tt < 6; tt += 4) {        // 6 col tiles of 16 (96 dims)
    const int n0 = tt * 16;
    const float* Vc = Vb + base * QSTR + hd * HDP + n0 + lrow;
    v8f acc = {};
#pragma unroll 8
    for (int k = 0; k < Nseq; k += 4) {
      v2f fa = *(const v2f*)(Ar + k);
      int r0 = k + 2 * half;
      v2f fb;
      fb.x = Vc[(size_t)r0 * QSTR];
      fb.y = Vc[(size_t)(r0 + 1) * QSTR];
      acc = wmma4(fa, fb, acc);
    }
#pragma unroll
    for (int r = 0; r < 8; ++r) {
      int qr = r + 8 * half;
      O[(base + q0 + qr) * QSTR + hd * HDP + n0 + lrow] = acc[r] * s_inv[qr];
    }
  }
}

// ---------------------------------------------------------------------------
// Host driver
// ---------------------------------------------------------------------------
extern "C" void kernel_launch(void* const* d_in, const int* in_sizes, int n_in,
                              void* d_out, int out_size, void* d_ws, size_t ws_size,
                              hipStream_t stream) {
  (void)in_sizes; (void)n_in; (void)out_size; (void)ws_size;
  const float* attn_bias = (const float*)d_in[0];
  const float* x         = (const float*)d_in[1];
  const int*   heights   = (const int*)d_in[2];
  const float* colE  = (const float*)d_in[3];
  const float* flnB  = (const float*)d_in[4];
  const float* flnG  = (const float*)d_in[5];
  const float* heE   = (const float*)d_in[6];
  struct Lp { const float *f1b,*f1w,*f2b,*f2w,*kb,*kw,*ln1b,*ln1g,*ln2b,*ln2g,
                          *ob,*ow,*qb,*qw,*vb,*vw; };
  Lp L[4];
  for (int l = 0; l < 4; ++l) {
    void* const* p = d_in + 7 + 16 * l;
    L[l].f1b = (const float*)p[0];  L[l].f1w = (const float*)p[1];
    L[l].f2b = (const float*)p[2];  L[l].f2w = (const float*)p[3];
    L[l].kb  = (const float*)p[4];  L[l].kw  = (const float*)p[5];
    L[l].ln1b= (const float*)p[6];  L[l].ln1g= (const float*)p[7];
    L[l].ln2b= (const float*)p[8];  L[l].ln2g= (const float*)p[9];
    L[l].ob  = (const float*)p[10]; L[l].ow  = (const float*)p[11];
    L[l].qb  = (const float*)p[12]; L[l].qw  = (const float*)p[13];
    L[l].vb  = (const float*)p[14]; L[l].vw  = (const float*)p[15];
  }
  const float* bF  = (const float*)d_in[71]; const float* WF  = (const float*)d_in[72];
  const float* bF2 = (const float*)d_in[73]; const float* WF2 = (const float*)d_in[74];
  const float* bJ1 = (const float*)d_in[75]; const float* WJ1 = (const float*)d_in[76];
  const float* bJ2 = (const float*)d_in[77]; const float* WJ2 = (const float*)d_in[78];
  const float* opE = (const float*)d_in[79]; const float* posE= (const float*)d_in[80];
  const float* bP  = (const float*)d_in[81]; const float* WP  = (const float*)d_in[82];
  const float* tabE= (const float*)d_in[83]; const float* typeE=(const float*)d_in[84];

  float* ws = (float*)d_ws;
  const size_t HBUF = (size_t)NTOK * HPAD;          // 8,388,608 floats
  const size_t QBUF = (size_t)NTOK * QSTR;          // 9,437,184 floats
  float* Hb = ws;
  float* Yb = Hb + HBUF;      // stride 288: LN output and attention output
  float* Qb = Yb + QBUF;
  float* Kb = Qb + QBUF;
  float* Vb = Kb + QBUF;
  float* WPAD = Vb + QBUF;
  const size_t LSTRIDE = 820576;                    // floats per layer block

  for (int l = 0; l < 4; ++l)
    pad_weights_kernel<<<3206, 256, 0, stream>>>(
        L[l].qw, L[l].kw, L[l].vw, L[l].ow, L[l].f1w, L[l].f2w,
        L[l].qb, L[l].kb, L[l].vb, L[l].ob, L[l].f2b, WPAD + LSTRIDE * l);

  embed_kernel<<<NTOK, 128, 0, stream>>>(x, heights, colE, opE, posE, typeE,
                                         tabE, heE, WF, bF, WF2, bF2,
                                         WJ1, bJ1, WJ2, bJ2, WP, bP, Hb);

  const dim3 gQKV(QSTR / 32, NTOK / 128);           // (9, 256)
  const dim3 gO(HPAD / 32, NTOK / 128);             // (8, 256)
  for (int l = 0; l < 4; ++l) {
    float* WB = WPAD + LSTRIDE * l;
    float* WQ = WB;            float* WK = WB + 73728;
    float* WV = WB + 147456;   float* WO = WB + 221184;
    float* W1 = WB + 294912;   float* W2 = WB + 557056;
    float* bq = WB + 819200;   float* bk = WB + 819488;
    float* bv = WB + 819776;   float* bo = WB + 820064;
    float* b2 = WB + 820320;

    ln_kernel<<<NTOK / 8, 256, 0, stream>>>(Hb, Yb, L[l].ln1g, L[l].ln1b);
    gemm_kernel<<<gQKV, 128, 0, stream>>>(Yb, WQ, bq, Qb, nullptr,
                                          QSTR, HPAD, QSTR);
    gemm_kernel<<<gQKV, 128, 0, stream>>>(Yb, WK, bk, Kb, nullptr,
                                          QSTR, HPAD, QSTR);
    gemm_kernel<<<gQKV, 128, 0, stream>>>(Yb, WV, bv, Vb, nullptr,
                                          QSTR, HPAD, QSTR);
    attn_kernel<<<dim3(Nseq / 16, NHEAD, Bsz), 128, 0, stream>>>(
        Qb, Kb, Vb, attn_bias, Yb);
    gemm_kernel<<<gO, 128, 0, stream>>>(Yb, WO, bo, Hb, Hb,
                                        HPAD, QSTR, QSTR);
    ln_kernel<<<NTOK / 8, 256, 0, stream>>>(Hb, Yb, L[l].ln2g, L[l].ln2b);
    ffn_kernel<<<NTOK / 32, 256, 0, stream>>>(Hb, Yb, W1, L[l].f1b, W2, b2);
  }
  final_ln_kernel<<<8, 256, 0, stream>>>(Hb, (float*)d_out, flnG, flnB);
}